// EMDLoss_11931419148401
// MI455X (gfx1250) — compile-verified
//
#include <hip/hip_runtime.h>
#include <hip/hip_bf16.h>
#include <math.h>

// Chamfer (one-sided) distance:
//   out = mean_{b,n} min_m || pred[b,n,:] - target[b,m,:] ||_2
//
// Strategy (MI455X / gfx1250, wave32):
//   min_m d^2 = |p|^2 - 2 * max_m (p.t - 0.5*|t|^2)
//   The inner score is a K=4 fp32 GEMM:
//     A row (m)  = (tx, ty, tz, -0.5*|t|^2)
//     B col (n)  = (px, py, pz, 1.0)
//   computed with V_WMMA_F32_16X16X4_F32, folded with v_max per tile.

typedef __attribute__((ext_vector_type(2))) float v2f;
typedef __attribute__((ext_vector_type(8))) float v8f;

#define BB 32
#define NN 4096
#define MM 4096
#define NT_M (MM / 16)   // 256 m-tiles per batch
#define NT_N (NN / 16)   // 256 n-tiles per batch
#define NPART (BB * NT_N)  // 8192 wave partials

// ---------------------------------------------------------------------------
// Pass 0: pack target into the exact A-operand VGPR layout of
// V_WMMA_F32_16X16X4_F32 (wave32):
//   lanes 0..15 : VGPR0,1 = K0,K1 = (x, y)        for row m = T*16 + lane
//   lanes 16..31: VGPR0,1 = K2,K3 = (z, -0.5|t|^2) for row m = T*16 + lane-16
// apack[((b*NT_M + T)*32 + lane)] is a float2 -> one b64 load per lane/tile.
// ---------------------------------------------------------------------------
__global__ void pack_target_kernel(const float* __restrict__ target,
                                   float2* __restrict__ apack) {
    int tid = blockIdx.x * blockDim.x + threadIdx.x;
    if (tid >= BB * MM) return;
    int b = tid / MM;
    int m = tid % MM;
    int T = m >> 4;
    int L = m & 15;
    const float* t = target + ((size_t)b * MM + m) * 3;
    float x = t[0], y = t[1], z = t[2];
    float h = -0.5f * (x * x + y * y + z * z);
    size_t base = ((size_t)b * NT_M + T) * 32;
    apack[base + L]      = make_float2(x, y);   // lanes 0..15  (K0,K1)
    apack[base + 16 + L] = make_float2(z, h);   // lanes 16..31 (K2,K3)
}

// ---------------------------------------------------------------------------
// Main: one wave handles one (b, n-tile of 16 pred points); loops over all
// 256 m-tiles with WMMA, folding max per lane, then reduces.
// blockDim = 256 (8 waves), grid = NPART/8 = 1024 blocks.
// ---------------------------------------------------------------------------
__global__ void chamfer_wmma_kernel(const float* __restrict__ pred,
                                    const float2* __restrict__ apack,
                                    float* __restrict__ partial) {
    const int lane = threadIdx.x & 31;
    const int wid  = (blockIdx.x * blockDim.x + threadIdx.x) >> 5;  // 0..NPART-1
    const int b    = wid / NT_N;
    const int nt   = wid % NT_N;

    // Per-lane pred point: n = nt*16 + (lane % 16)
    const int nl = nt * 16 + (lane & 15);
    const float* pp = pred + ((size_t)b * NN + nl) * 3;
    const float px = pp[0], py = pp[1], pz = pp[2];
    const float p2 = px * px + py * py + pz * pz;

    // B operand (K=4 x 16 cols): lanes 0..15 -> (K0,K1)=(px,py),
    //                            lanes 16..31 -> (K2,K3)=(pz, 1.0)
    v2f bop;
    if (lane < 16) { bop[0] = px; bop[1] = py; }
    else           { bop[0] = pz; bop[1] = 1.0f; }

    v8f acc;
    #pragma unroll
    for (int i = 0; i < 8; ++i) acc[i] = -3.0e38f;

    const float2* ap = apack + (size_t)b * NT_M * 32;
    const v8f czero = {};

    #pragma unroll 4
    for (int T = 0; T < NT_M; ++T) {
        float2 a2 = ap[(size_t)T * 32 + lane];
        v2f aop; aop[0] = a2.x; aop[1] = a2.y;
        // D = A(16x4 target) x B(4x16 pred) + 0  ->  score = p.t - 0.5|t|^2
        v8f d = __builtin_amdgcn_wmma_f32_16x16x4_f32(
            /*neg_a=*/false, aop, /*neg_b=*/false, bop,
            /*c_mod=*/(short)0, czero, /*reuse_a=*/false, /*reuse_b=*/false);
        #pragma unroll
        for (int i = 0; i < 8; ++i) acc[i] = fmaxf(acc[i], d[i]);
    }

    // In-lane max over the 8 m-rows this lane holds.
    float s = acc[0];
    #pragma unroll
    for (int i = 1; i < 8; ++i) s = fmaxf(s, acc[i]);
    // Combine the two lane-halves (rows m..m+7 with rows m+8..m+15).
    s = fmaxf(s, __shfl_xor(s, 16, 32));

    // min_m d^2 = p2 - 2*max score ; distance via one sqrt per point.
    float dist = sqrtf(fmaxf(p2 - 2.0f * s, 0.0f));

    // Wave sum: lanes L and L+16 hold duplicates -> sum all 32, halve.
    #pragma unroll
    for (int off = 16; off >= 1; off >>= 1)
        dist += __shfl_xor(dist, off, 32);
    if (lane == 0) partial[wid] = dist * 0.5f;
}

// ---------------------------------------------------------------------------
// Final deterministic tree reduction of NPART partials -> mean.
// Single block of 256 threads.
// ---------------------------------------------------------------------------
__global__ void final_reduce_kernel(const float* __restrict__ partial,
                                    float* __restrict__ out) {
    __shared__ float sh[256];
    float s = 0.0f;
    for (int i = threadIdx.x; i < NPART; i += 256) s += partial[i];
    sh[threadIdx.x] = s;
    __syncthreads();
    for (int stride = 128; stride >= 1; stride >>= 1) {
        if (threadIdx.x < stride) sh[threadIdx.x] += sh[threadIdx.x + stride];
        __syncthreads();
    }
    if (threadIdx.x == 0) out[0] = sh[0] / (float)(BB * NN);
}

extern "C" void kernel_launch(void* const* d_in, const int* in_sizes, int n_in,
                              void* d_out, int out_size, void* d_ws, size_t ws_size,
                              hipStream_t stream) {
    const float* pred   = (const float*)d_in[0];   // [B, N, 3] f32
    const float* target = (const float*)d_in[1];   // [B, M, 3] f32
    float* out = (float*)d_out;                    // scalar f32

    // Workspace layout: apack (2 MB) | partials (32 KB)
    float2* apack  = (float2*)d_ws;
    float* partial = (float*)((char*)d_ws + (size_t)BB * NT_M * 32 * sizeof(float2));

    // Pass 0: pack target tiles into WMMA A-operand layout.
    {
        int threads = 256;
        int blocks = (BB * MM + threads - 1) / threads;  // 512
        pack_target_kernel<<<blocks, threads, 0, stream>>>(target, apack);
    }
    // Main: 8192 waves, 8 waves per block.
    {
        int threads = 256;
        int blocks = NPART / 8;  // 1024
        chamfer_wmma_kernel<<<blocks, threads, 0, stream>>>(pred, apack, partial);
    }
    // Final mean.
    final_reduce_kernel<<<1, 256, 0, stream>>>(partial, out);
}